// model_audio_40089224741244
// MI455X (gfx1250) — compile-verified
//
#include <hip/hip_runtime.h>
#include <math.h>

// ---- problem dimensions -------------------------------------------------
#define BB    16      // batch
#define L0C   1021
#define L1C   509     // after conv1 stride2 k5
#define L1P   254     // after pool
#define L2C   250     // after conv4 k5
#define L2P   125
#define L3C   123     // after conv5 k3
#define CA    5       // attention channels
#define NNA   3936    // 123*32
#define NFLAT 19680   // 5*3936
#define NH    1024
#define NCLS  13
// attention K/V LDS chunking: 3936 = 8 * 492
#define MCH   492
#define NCHK  8
// FC1 K split: 19680 = 120 * 164, 164 = 41 * 4
#define KCHUNK 164
#define KSPLIT 120
#define KSTEPS 41

typedef __attribute__((ext_vector_type(2))) float v2f;
typedef __attribute__((ext_vector_type(8))) float v8f;

__device__ __forceinline__ float eluf(float x) {
    return x > 0.f ? x : (__expf(x) - 1.f);
}

// ---- generic 1D conv: dst[b,co,l] = bias + sum_{ci,k} src[b,ci,l*stride+k]*w[co,ci,k]
__global__ void conv1d_kernel(const float* __restrict__ src, const float* __restrict__ w,
                              const float* __restrict__ bias, float* __restrict__ dst,
                              int Cin, int Cout, int Lin, int Lout, int K, int stride, int n)
{
    int idx = blockIdx.x * blockDim.x + threadIdx.x;
    if (idx >= n) return;
    int l  = idx % Lout;
    int t  = idx / Lout;
    int co = t % Cout;
    int b  = t / Cout;
    const float* sp = src + (size_t)b * Cin * Lin + l * stride;
    const float* wp = w + (size_t)co * Cin * K;
    float acc = bias[co];
    for (int ci = 0; ci < Cin; ++ci) {
        const float* s  = sp + ci * Lin;
        const float* ww = wp + ci * K;
        for (int kk = 0; kk < K; ++kk) acc += s[kk] * ww[kk];
    }
    dst[idx] = acc;
}

// ---- per-channel mean/var over (batch, length); one block per channel ----
__global__ void chstats_kernel(const float* __restrict__ src, float* __restrict__ meanp,
                               float* __restrict__ varp, int C, int L)
{
    __shared__ float s1[256];
    __shared__ float s2[256];
    int c = blockIdx.x;
    int t = threadIdx.x;
    float sum = 0.f, sq = 0.f;
    int tot = BB * L;
    for (int i = t; i < tot; i += 256) {
        int b = i / L, l = i % L;
        float v = src[(size_t)(b * C + c) * L + l];
        sum += v; sq += v * v;
    }
    s1[t] = sum; s2[t] = sq;
    __syncthreads();
    for (int s = 128; s > 0; s >>= 1) {
        if (t < s) { s1[t] += s1[t + s]; s2[t] += s2[t + s]; }
        __syncthreads();
    }
    if (t == 0) {
        float m = s1[0] / (float)tot;
        meanp[c] = m;
        varp[c]  = s2[0] / (float)tot - m * m;
    }
}

// ---- BN + ELU + maxpool(2,2) --------------------------------------------
__global__ void bnelupool_kernel(const float* __restrict__ src, const float* __restrict__ meanp,
                                 const float* __restrict__ varp, const float* __restrict__ g,
                                 const float* __restrict__ bt, float* __restrict__ dst,
                                 int C, int Lin, int Lout, int n)
{
    int idx = blockIdx.x * blockDim.x + threadIdx.x;
    if (idx >= n) return;
    int i = idx % Lout;
    int t = idx / Lout;
    int c = t % C;
    int b = t / C;
    float m   = meanp[c];
    float inv = rsqrtf(varp[c] + 1e-5f);
    float ga = g[c], be = bt[c];
    const float* s = src + (size_t)(b * C + c) * Lin + 2 * i;
    float v0 = eluf((s[0] - m) * inv * ga + be);
    float v1 = eluf((s[1] - m) * inv * ga + be);
    dst[idx] = fmaxf(v0, v1);
}

// ---- BN + ELU + transpose to feat[b, l*32 + c] (src is (B,32,123)) ------
__global__ void bnelufeat_kernel(const float* __restrict__ src, const float* __restrict__ meanp,
                                 const float* __restrict__ varp, const float* __restrict__ g,
                                 const float* __restrict__ bt, float* __restrict__ feat, int n)
{
    int idx = blockIdx.x * blockDim.x + threadIdx.x;
    if (idx >= n) return;
    int l = idx % L3C;
    int c = (idx / L3C) % 32;
    int b = idx / (L3C * 32);
    float m   = meanp[c];
    float inv = rsqrtf(varp[c] + 1e-5f);
    float v = eluf((src[idx] - m) * inv * g[c] + bt[c]);
    feat[(size_t)b * NNA + l * 32 + c] = v;
}

// ---- broadcast feat across 5 channels: x1[b,c,n] = feat[b,n] ------------
__global__ void bcast_kernel(const float* __restrict__ feat, float* __restrict__ x1, int n)
{
    int idx = blockIdx.x * blockDim.x + threadIdx.x;
    if (idx >= n) return;
    int nn = idx % NNA;
    int b  = idx / (CA * NNA);
    x1[idx] = feat[(size_t)b * NNA + nn];
}

// ---- q,k,v = 1x1 conv over channel dim (5x5 matrices) -------------------
__global__ void qkv_kernel(const float* __restrict__ x,
                           const float* __restrict__ wq, const float* __restrict__ bq,
                           const float* __restrict__ wk, const float* __restrict__ bk,
                           const float* __restrict__ wv, const float* __restrict__ bv,
                           float* __restrict__ q, float* __restrict__ k, float* __restrict__ v,
                           int n)
{
    int idx = blockIdx.x * blockDim.x + threadIdx.x;
    if (idx >= n) return;
    int nn = idx % NNA;
    int b  = idx / NNA;
    float xv[CA];
#pragma unroll
    for (int j = 0; j < CA; ++j) xv[j] = x[(size_t)(b * CA + j) * NNA + nn];
#pragma unroll
    for (int c = 0; c < CA; ++c) {
        float aq = bq[c], ak = bk[c], av = bv[c];
#pragma unroll
        for (int j = 0; j < CA; ++j) {
            aq += wq[c * CA + j] * xv[j];
            ak += wk[c * CA + j] * xv[j];
            av += wv[c * CA + j] * xv[j];
        }
        q[(size_t)(b * CA + c) * NNA + nn] = aq;
        k[(size_t)(b * CA + c) * NNA + nn] = ak;
        v[(size_t)(b * CA + c) * NNA + nn] = av;
    }
}

// ---- streaming softmax attention: out = gamma*softmax(q^T k) applied to v + x
__global__ void attn_kernel(const float* __restrict__ x, const float* __restrict__ q,
                            const float* __restrict__ k, const float* __restrict__ v,
                            const float* __restrict__ gammap, float* __restrict__ out)
{
    __shared__ float lk[CA * MCH];
    __shared__ float lv[CA * MCH];
    int b = blockIdx.y;
    int t = threadIdx.x;
    int nn = blockIdx.x * 256 + t;
    bool act = nn < NNA;

    float qr[CA] = {0.f, 0.f, 0.f, 0.f, 0.f};
    if (act) {
#pragma unroll
        for (int c = 0; c < CA; ++c) qr[c] = q[(size_t)(b * CA + c) * NNA + nn];
    }
    float mx = -INFINITY, ssum = 0.f;
    float acc[CA] = {0.f, 0.f, 0.f, 0.f, 0.f};

    for (int ch = 0; ch < NCHK; ++ch) {
        int mb = ch * MCH;
        for (int i = t; i < CA * MCH; i += 256) {
            int c = i / MCH, m = i % MCH;
            lk[i] = k[(size_t)(b * CA + c) * NNA + mb + m];
            lv[i] = v[(size_t)(b * CA + c) * NNA + mb + m];
        }
        __syncthreads();
        if (act) {
            for (int m = 0; m < MCH; ++m) {
                float e = qr[0] * lk[m] + qr[1] * lk[MCH + m] + qr[2] * lk[2 * MCH + m]
                        + qr[3] * lk[3 * MCH + m] + qr[4] * lk[4 * MCH + m];
                if (e > mx) {
                    float corr = __expf(mx - e);
                    ssum *= corr;
#pragma unroll
                    for (int c = 0; c < CA; ++c) acc[c] *= corr;
                    mx = e;
                }
                float wgt = __expf(e - mx);
                ssum += wgt;
#pragma unroll
                for (int c = 0; c < CA; ++c) acc[c] += wgt * lv[c * MCH + m];
            }
        }
        __syncthreads();
    }
    if (act) {
        float g   = gammap[0];
        float inv = 1.f / ssum;
#pragma unroll
        for (int c = 0; c < CA; ++c) {
            size_t o = (size_t)(b * CA + c) * NNA + nn;
            out[o] = g * (acc[c] * inv) + x[o];
        }
    }
}

// ---- transpose last two dims (123,32)->(32,123) per (b,c) ---------------
__global__ void transp_kernel(const float* __restrict__ src, float* __restrict__ dst, int n)
{
    int idx = blockIdx.x * blockDim.x + threadIdx.x;
    if (idx >= n) return;
    int n2 = idx % NNA;
    int bc = idx / NNA;
    int h = n2 / L3C;
    int w = n2 % L3C;
    dst[idx] = src[(size_t)bc * NNA + w * 32 + h];
}

// ---- FC1: (16 x 19680) @ (19680 x 1024) via V_WMMA_F32_16X16X4_F32 ------
// one wave per block; grid = (64 N-tiles, 120 K-splits); fixed-order reduce after
__global__ void fc1_wmma_kernel(const float* __restrict__ flat, const float* __restrict__ w1,
                                float* __restrict__ partial)
{
    int lane = threadIdx.x;          // 0..31
    int nb   = blockIdx.x;           // 0..63  -> N tile
    int ks   = blockIdx.y;           // 0..119 -> K split
    int m    = lane & 15;
    int half = lane >> 4;

    // A fragment: lane holds A[M=m, K=2*half+{0,1}]  (flat is row-major (16,19680))
    const float* Ar = flat + (size_t)m * NFLAT;
    // B fragment: lane holds B[K=2*half+{0,1}, N=m] = fc1_w[nb*16+m, K] (B = W^T)
    const float* Br = w1 + (size_t)(nb * 16 + m) * NFLAT;

    v8f acc = {0.f, 0.f, 0.f, 0.f, 0.f, 0.f, 0.f, 0.f};
    int kb = ks * KCHUNK + half * 2;
#pragma unroll 4
    for (int s = 0; s < KSTEPS; ++s) {
        int kk = kb + s * 4;
        v2f a  = *(const v2f*)(Ar + kk);
        v2f bf = *(const v2f*)(Br + kk);
        acc = __builtin_amdgcn_wmma_f32_16x16x4_f32(false, a, false, bf,
                                                    (short)0, acc, false, false);
    }
    // D layout: VGPR r, lane l -> M = r + 8*(l>>4), N = l&15
    float* outp = partial + (size_t)ks * (16 * NH) + nb * 16 + m;
#pragma unroll
    for (int r = 0; r < 8; ++r)
        outp[(size_t)(r + 8 * half) * NH] = acc[r];
}

__global__ void fc1_reduce_kernel(const float* __restrict__ partial,
                                  const float* __restrict__ bias, float* __restrict__ h)
{
    int idx = blockIdx.x * 256 + threadIdx.x;   // 0..16383
    if (idx >= 16 * NH) return;
    int o = idx % NH;
    float s = bias[o];
    for (int ks = 0; ks < KSPLIT; ++ks) s += partial[(size_t)ks * (16 * NH) + idx];
    h[idx] = s;
}

// ---- BN over batch (axis=1 features) + ELU, in place --------------------
__global__ void bn6_elu_kernel(float* __restrict__ h, const float* __restrict__ g,
                               const float* __restrict__ bt)
{
    int o = blockIdx.x * 256 + threadIdx.x;
    if (o >= NH) return;
    float s = 0.f, sq = 0.f;
    for (int b = 0; b < BB; ++b) {
        float v = h[(size_t)b * NH + o];
        s += v; sq += v * v;
    }
    float m   = s / (float)BB;
    float var = sq / (float)BB - m * m;
    float inv = rsqrtf(var + 1e-5f);
    float ga = g[o], be = bt[o];
    for (int b = 0; b < BB; ++b) {
        float v = (h[(size_t)b * NH + o] - m) * inv * ga + be;
        h[(size_t)b * NH + o] = eluf(v);
    }
}

// ---- FC2 (13 outputs) + softmax; one block per batch row ----------------
__global__ void fc2_softmax_kernel(const float* __restrict__ h, const float* __restrict__ w2,
                                   const float* __restrict__ b2, float* __restrict__ out)
{
    __shared__ float red[NCLS * 256];
    int b = blockIdx.x;
    int t = threadIdx.x;
    float p[NCLS];
#pragma unroll
    for (int o = 0; o < NCLS; ++o) p[o] = 0.f;
    for (int k = t; k < NH; k += 256) {
        float xv = h[(size_t)b * NH + k];
#pragma unroll
        for (int o = 0; o < NCLS; ++o) p[o] += xv * w2[(size_t)o * NH + k];
    }
#pragma unroll
    for (int o = 0; o < NCLS; ++o) red[o * 256 + t] = p[o];
    __syncthreads();
    for (int s = 128; s > 0; s >>= 1) {
        if (t < s) {
#pragma unroll
            for (int o = 0; o < NCLS; ++o) red[o * 256 + t] += red[o * 256 + t + s];
        }
        __syncthreads();
    }
    if (t == 0) {
        float lg[NCLS];
        float mx = -INFINITY;
#pragma unroll
        for (int o = 0; o < NCLS; ++o) { lg[o] = red[o * 256] + b2[o]; mx = fmaxf(mx, lg[o]); }
        float ss = 0.f;
#pragma unroll
        for (int o = 0; o < NCLS; ++o) { lg[o] = __expf(lg[o] - mx); ss += lg[o]; }
        float inv = 1.f / ss;
#pragma unroll
        for (int o = 0; o < NCLS; ++o) out[b * NCLS + o] = lg[o] * inv;
    }
}

// -------------------------------------------------------------------------
extern "C" void kernel_launch(void* const* d_in, const int* in_sizes, int n_in,
                              void* d_out, int out_size, void* d_ws, size_t ws_size,
                              hipStream_t stream)
{
    const float* x     = (const float*)d_in[0];
    const float* c1_w  = (const float*)d_in[1];
    const float* c1_b  = (const float*)d_in[2];
    const float* bn1_g = (const float*)d_in[3];
    const float* bn1_b = (const float*)d_in[4];
    const float* c4_w  = (const float*)d_in[5];
    const float* c4_b  = (const float*)d_in[6];
    const float* bn4_g = (const float*)d_in[7];
    const float* bn4_b = (const float*)d_in[8];
    const float* c5_w  = (const float*)d_in[9];
    const float* c5_b  = (const float*)d_in[10];
    const float* bn5_g = (const float*)d_in[11];
    const float* bn5_b = (const float*)d_in[12];
    const float* a1_wq = (const float*)d_in[13];
    const float* a1_bq = (const float*)d_in[14];
    const float* a1_wk = (const float*)d_in[15];
    const float* a1_bk = (const float*)d_in[16];
    const float* a1_wv = (const float*)d_in[17];
    const float* a1_bv = (const float*)d_in[18];
    const float* a1_g  = (const float*)d_in[19];
    const float* a2_wq = (const float*)d_in[20];
    const float* a2_bq = (const float*)d_in[21];
    const float* a2_wk = (const float*)d_in[22];
    const float* a2_bk = (const float*)d_in[23];
    const float* a2_wv = (const float*)d_in[24];
    const float* a2_bv = (const float*)d_in[25];
    const float* a2_g  = (const float*)d_in[26];
    const float* fc1_w = (const float*)d_in[27];
    const float* fc1_b = (const float*)d_in[28];
    const float* bn6_g = (const float*)d_in[29];
    const float* bn6_b = (const float*)d_in[30];
    const float* fc2_w = (const float*)d_in[31];
    const float* fc2_b = (const float*)d_in[32];
    float* out = (float*)d_out;

    // ---- workspace layout (floats) ----
    float* ws    = (float*)d_ws;
    float* y1    = ws;                 // 16*32*509 = 260608
    float* p2    = y1 + 260608;        // 16*32*254 = 130048
    float* y3    = p2 + 130048;        // 16*64*250 = 256000
    float* p4    = y3 + 256000;        // 16*64*125 = 128000
    float* y5    = p4 + 128000;        // 16*32*123 = 62976
    float* feat  = y5 + 62976;         // 62976
    float* x1    = feat + 62976;       // 16*5*3936 = 314880
    float* qb    = x1 + 314880;
    float* kb    = qb + 314880;
    float* vb    = kb + 314880;
    float* ya    = vb + 314880;
    float* x2    = ya + 314880;
    float* flat  = x2 + 314880;        // doubles as attn2 output == FC input
    float* part  = flat + 314880;      // 120*16*1024 = 1966080
    float* hbuf  = part + 1966080;     // 16*1024
    float* mean1 = hbuf + 16384;       // 64
    float* var1  = mean1 + 64;         // 64

    const int T = 256;
    auto blks = [](int n, int t) { return (n + t - 1) / t; };

    // stage 1: conv1 (stride 2) -> bn -> elu -> pool
    conv1d_kernel<<<blks(BB*32*L1C, T), T, 0, stream>>>(x, c1_w, c1_b, y1, 5, 32, L0C, L1C, 5, 2, BB*32*L1C);
    chstats_kernel<<<32, T, 0, stream>>>(y1, mean1, var1, 32, L1C);
    bnelupool_kernel<<<blks(BB*32*L1P, T), T, 0, stream>>>(y1, mean1, var1, bn1_g, bn1_b, p2, 32, L1C, L1P, BB*32*L1P);
    // stage 2: conv4 -> bn -> elu -> pool
    conv1d_kernel<<<blks(BB*64*L2C, T), T, 0, stream>>>(p2, c4_w, c4_b, y3, 32, 64, L1P, L2C, 5, 1, BB*64*L2C);
    chstats_kernel<<<64, T, 0, stream>>>(y3, mean1, var1, 64, L2C);
    bnelupool_kernel<<<blks(BB*64*L2P, T), T, 0, stream>>>(y3, mean1, var1, bn4_g, bn4_b, p4, 64, L2C, L2P, BB*64*L2P);
    // stage 3: conv5 -> bn -> elu -> transpose to feat
    conv1d_kernel<<<blks(BB*32*L3C, T), T, 0, stream>>>(p4, c5_w, c5_b, y5, 64, 32, L2P, L3C, 3, 1, BB*32*L3C);
    chstats_kernel<<<32, T, 0, stream>>>(y5, mean1, var1, 32, L3C);
    bnelufeat_kernel<<<blks(BB*32*L3C, T), T, 0, stream>>>(y5, mean1, var1, bn5_g, bn5_b, feat, BB*32*L3C);

    // attention 1 (on broadcast over 5 channels, N = 123*32)
    bcast_kernel<<<blks(BB*CA*NNA, T), T, 0, stream>>>(feat, x1, BB*CA*NNA);
    qkv_kernel<<<blks(BB*NNA, T), T, 0, stream>>>(x1, a1_wq, a1_bq, a1_wk, a1_bk, a1_wv, a1_bv, qb, kb, vb, BB*NNA);
    attn_kernel<<<dim3(16, BB), T, 0, stream>>>(x1, qb, kb, vb, a1_g, ya);

    // attention 2 (on transposed (32,123) layout); its output IS the FC flat input
    transp_kernel<<<blks(BB*CA*NNA, T), T, 0, stream>>>(ya, x2, BB*CA*NNA);
    qkv_kernel<<<blks(BB*NNA, T), T, 0, stream>>>(x2, a2_wq, a2_bq, a2_wk, a2_bk, a2_wv, a2_bv, qb, kb, vb, BB*NNA);
    attn_kernel<<<dim3(16, BB), T, 0, stream>>>(x2, qb, kb, vb, a2_g, flat);

    // FC1 via fp32 WMMA, K-split + deterministic reduce
    fc1_wmma_kernel<<<dim3(64, KSPLIT), 32, 0, stream>>>(flat, fc1_w, part);
    fc1_reduce_kernel<<<blks(BB*NH, T), T, 0, stream>>>(part, fc1_b, hbuf);
    bn6_elu_kernel<<<blks(NH, T), T, 0, stream>>>(hbuf, bn6_g, bn6_b);
    fc2_softmax_kernel<<<BB, T, 0, stream>>>(hbuf, fc2_w, fc2_b, out);

    (void)in_sizes; (void)n_in; (void)out_size; (void)ws_size;
}